// DropoutSoftmax_22917945491859
// MI455X (gfx1250) — compile-verified
//
#include <hip/hip_runtime.h>
#include <stdint.h>

// Problem constants (match reference.py)
#define ROWS 32768
#define COLS 2048
#define KDIM 512
#define EPS  1e-5f

#define WAVES_PER_BLOCK 8          // 256 threads = 8 wave32
#define BLOCK_THREADS   (WAVES_PER_BLOCK * 32)

// One wave32 handles one row:
//  - async-DMA the full 2048-float row into an LDS slice (coalesced b128);
//    the offset: immediate advances BOTH the global and LDS address, so a
//    single base address pair serves all 16 DMA instructions
//  - coalesced b128 loads of idx / gamma / beta
//  - LDS gather (randomness absorbed by 64 LDS banks)
//  - layernorm + softmax with wave32 shuffle allreduces
//  - rebuild output row in the same LDS slice (zero + scatter)
//  - async-DMA LDS -> global (coalesced b128), no random global stores

#define ALOAD(OFF) \
    asm volatile("global_load_async_to_lds_b128 %0, %1, off offset:" #OFF \
                 :: "v"(la), "v"(ga) : "memory")
#define ASTORE(OFF) \
    asm volatile("global_store_async_from_lds_b128 %0, %1, off offset:" #OFF \
                 :: "v"(gs), "v"(ls) : "memory")

__global__ __launch_bounds__(BLOCK_THREADS, 1)
void ln_softmax_gather_scatter(const float* __restrict__ x,
                               const float* __restrict__ gamma,
                               const float* __restrict__ beta,
                               const int*   __restrict__ cidx,
                               float*       __restrict__ out)
{
    __shared__ __align__(16) float lds[WAVES_PER_BLOCK][COLS];

    const int lane = threadIdx.x & 31;
    const int wave = threadIdx.x >> 5;
    const int row  = blockIdx.x * WAVES_PER_BLOCK + wave;

    const float* xrow = x   + (size_t)row * COLS;
    float*       orow = out + (size_t)row * COLS;

    // Low 32 bits of the generic (flat) LDS pointer == raw LDS byte offset
    // (flat LDS addresses are {SHARED_BASE, offset32} per CDNA5 aperture rules).
    const unsigned ldsbase = (unsigned)(uintptr_t)(&lds[wave][0]);

    // ---- 1) Stage the full row into LDS: CDNA5 async global->LDS b128 ----
    // Lane L moves bytes [L*16, L*16+16) of each 512-B chunk; the offset:
    // immediate selects the chunk on both the global and LDS side.
    {
        unsigned long long ga = (unsigned long long)(uintptr_t)(xrow + lane * 4);
        unsigned           la = ldsbase + (unsigned)(lane * 16);
        ALOAD(0);    ALOAD(512);  ALOAD(1024); ALOAD(1536);
        ALOAD(2048); ALOAD(2560); ALOAD(3072); ALOAD(3584);
        ALOAD(4096); ALOAD(4608); ALOAD(5120); ALOAD(5632);
        ALOAD(6144); ALOAD(6656); ALOAD(7168); ALOAD(7680);
    }

    // ---- 2) Overlap: coalesced b128 loads of indices / gamma / beta ----
    // Lane owns k-positions 4*lane+128*j+c  (j=0..3, c=0..3)
    const int4*   idx4 = (const int4*)(cidx + (size_t)row * KDIM);
    const float4* g4   = (const float4*)gamma;
    const float4* b4   = (const float4*)beta;
    int4   iv[4];
    float4 gv[4], bv[4];
#pragma unroll
    for (int j = 0; j < 4; ++j) {
        iv[j] = idx4[lane + 32 * j];
        gv[j] = g4[lane + 32 * j];
        bv[j] = b4[lane + 32 * j];
    }

    // ---- 3) Wait for the DMA, then gather from LDS ----
    asm volatile("s_wait_asynccnt 0x0" ::: "memory");

    float v[16];
#pragma unroll
    for (int j = 0; j < 4; ++j) {
        const int4 I = iv[j];
        v[4 * j + 0] = lds[wave][I.x];
        v[4 * j + 1] = lds[wave][I.y];
        v[4 * j + 2] = lds[wave][I.z];
        v[4 * j + 3] = lds[wave][I.w];
    }

    // ---- 4) LayerNorm: one-pass sum / sum-of-squares + wave32 allreduce ----
    float s = 0.0f, ss = 0.0f;
#pragma unroll
    for (int t = 0; t < 16; ++t) { s += v[t]; ss += v[t] * v[t]; }
#pragma unroll
    for (int m = 16; m >= 1; m >>= 1) {
        s  += __shfl_xor(s,  m, 32);
        ss += __shfl_xor(ss, m, 32);
    }
    const float mu   = s * (1.0f / (float)KDIM);
    const float var  = fmaxf(ss * (1.0f / (float)KDIM) - mu * mu, 0.0f);
    const float rstd = rsqrtf(var + EPS);

    // normalize + affine, track running max for softmax
    float mx = -3.402823466e38f;
#pragma unroll
    for (int j = 0; j < 4; ++j) {
        const float4 G = gv[j], B = bv[j];
        v[4 * j + 0] = (v[4 * j + 0] - mu) * rstd * G.x + B.x;
        v[4 * j + 1] = (v[4 * j + 1] - mu) * rstd * G.y + B.y;
        v[4 * j + 2] = (v[4 * j + 2] - mu) * rstd * G.z + B.z;
        v[4 * j + 3] = (v[4 * j + 3] - mu) * rstd * G.w + B.w;
    }
#pragma unroll
    for (int t = 0; t < 16; ++t) mx = fmaxf(mx, v[t]);
#pragma unroll
    for (int m = 16; m >= 1; m >>= 1) mx = fmaxf(mx, __shfl_xor(mx, m, 32));

    // ---- 5) Softmax ----
    float es = 0.0f;
#pragma unroll
    for (int t = 0; t < 16; ++t) { v[t] = __expf(v[t] - mx); es += v[t]; }
#pragma unroll
    for (int m = 16; m >= 1; m >>= 1) es += __shfl_xor(es, m, 32);
    const float inv = 1.0f / es;

    // ---- 6) Rebuild output row in the same LDS slice ----
    // Same-wave DS ops are in-order: gather-loads above already consumed,
    // zero-fill then scatter are ordered stores.
    asm volatile("" ::: "memory");
    float4* s4 = (float4*)(&lds[wave][0]);
    const float4 z = make_float4(0.0f, 0.0f, 0.0f, 0.0f);
#pragma unroll
    for (int j = 0; j < COLS / 128; ++j) s4[j * 32 + lane] = z;
#pragma unroll
    for (int j = 0; j < 4; ++j) {
        const int4 I = iv[j];
        lds[wave][I.x] = v[4 * j + 0] * inv;
        lds[wave][I.y] = v[4 * j + 1] * inv;
        lds[wave][I.z] = v[4 * j + 2] * inv;
        lds[wave][I.w] = v[4 * j + 3] * inv;
    }

    // ---- 7) Drain DS, then async LDS -> global coalesced b128 stores ----
    // (DScnt and ASYNCcnt are independent counters: must fence DS before the
    //  async engine reads LDS. S_ENDPGM's implicit wait-idle covers drain.)
    asm volatile("s_wait_dscnt 0x0" ::: "memory");
    {
        unsigned long long gs = (unsigned long long)(uintptr_t)(orow + lane * 4);
        unsigned           ls = ldsbase + (unsigned)(lane * 16);
        ASTORE(0);    ASTORE(512);  ASTORE(1024); ASTORE(1536);
        ASTORE(2048); ASTORE(2560); ASTORE(3072); ASTORE(3584);
        ASTORE(4096); ASTORE(4608); ASTORE(5120); ASTORE(5632);
        ASTORE(6144); ASTORE(6656); ASTORE(7168); ASTORE(7680);
    }
}

extern "C" void kernel_launch(void* const* d_in, const int* in_sizes, int n_in,
                              void* d_out, int out_size, void* d_ws, size_t ws_size,
                              hipStream_t stream) {
    const float* x     = (const float*)d_in[0];   // [2,16,1024,2048] f32
    const float* gamma = (const float*)d_in[1];   // [512] f32
    const float* beta  = (const float*)d_in[2];   // [512] f32
    const int*   cidx  = (const int*)d_in[3];     // [32768,512] i32
    float*       out   = (float*)d_out;           // [2,16,1024,2048] f32

    const int rows   = in_sizes[3] / KDIM;        // 32768
    const int blocks = rows / WAVES_PER_BLOCK;    // 4096

    ln_softmax_gather_scatter<<<blocks, BLOCK_THREADS, 0, stream>>>(
        x, gamma, beta, cidx, out);
}